// QKProjectionLayer_1391569404500
// MI455X (gfx1250) — compile-verified
//
#include <hip/hip_runtime.h>
#include <math.h>

// Problem constants (match reference)
constexpr int B  = 4;
constexpr int L  = 2048;
constexpr int D  = 256;
constexpr int C  = 64;         // chunk length
constexpr int NC = L / C;      // 32 chunks per batch
constexpr float EPS = 1e-7f;

typedef float v2f __attribute__((ext_vector_type(2)));
typedef float v8f __attribute__((ext_vector_type(8)));

// D(16x16 f32) = A(16x4 f32) x B(4x16 f32) + C   -- CDNA5 V_WMMA_F32_16X16X4_F32
static __device__ __forceinline__ v8f wmma_f32(v2f a, v2f b, v8f c) {
  return __builtin_amdgcn_wmma_f32_16x16x4_f32(false, a, false, b, (short)0, c,
                                               false, false);
}

// ---------------------------------------------------------------------------
// Kernel 1: per-(batch,chunk) Gram matrices  M_c = K_c^T K_c   [D x D]
// grid = B*NC workgroups, 256 threads (8 waves). Each wave: 32 output tiles.
// ---------------------------------------------------------------------------
__global__ __launch_bounds__(256) void gram_kernel(const float* __restrict__ kin,
                                                   float* __restrict__ M) {
  const int bc = blockIdx.x;                       // b*NC + c ; chunk start = bc*C*D
  const float* Kc = kin + (size_t)bc * C * D;      // [C x D] row-major
  float* Mout     = M   + (size_t)bc * D * D;

  __shared__ float sK[C][D];                       // 64 KB
  const int tid = threadIdx.x;
  for (int i = tid; i < C * D; i += 256) sK[i >> 8][i & 255] = Kc[i];
  __syncthreads();

  const int wave = tid >> 5, lane = tid & 31;
  const int lo = lane & 15, hi = lane >> 4;

  for (int t = 0; t < 32; ++t) {
    const int tau = wave * 32 + t;                 // 0..255 tiles of D/16 x D/16
    const int mi = tau & 15, ni = tau >> 4;
    v8f acc = {};
    for (int k0 = 0; k0 < C; k0 += 4) {
      v2f a, b;
      a.x = sK[k0 + 2 * hi][mi * 16 + lo];         // A[m,k] = K[t=k, d=mi*16+m]
      a.y = sK[k0 + 2 * hi + 1][mi * 16 + lo];
      b.x = sK[k0 + 2 * hi][ni * 16 + lo];         // B[k,n] = K[t=k, d=ni*16+n]
      b.y = sK[k0 + 2 * hi + 1][ni * 16 + lo];
      acc = wmma_f32(a, b, acc);
    }
    for (int r = 0; r < 8; ++r)
      Mout[(size_t)(mi * 16 + r + 8 * hi) * D + ni * 16 + lo] = acc[r];
  }
}

// ---------------------------------------------------------------------------
// Kernel 2: in-place exclusive prefix over chunks: M[c] <- P_prev + sum_{c'<c} M[c']
// and write P_final (inclusive sum) to output. One thread per (b, element).
// ---------------------------------------------------------------------------
__global__ __launch_bounds__(256) void scan_kernel(const float* __restrict__ Pprev,
                                                   float* __restrict__ M,
                                                   float* __restrict__ Pfinal) {
  const size_t idx = (size_t)blockIdx.x * 256 + threadIdx.x;   // 0 .. B*D*D-1
  const int b = (int)(idx / (size_t)(D * D));
  const size_t e = idx % (size_t)(D * D);
  float acc = Pprev[idx];
  float* p = M + ((size_t)b * NC) * (size_t)(D * D) + e;
  for (int c = 0; c < NC; ++c) {
    const float tmp = p[(size_t)c * D * D];
    p[(size_t)c * D * D] = acc;                    // exclusive prefix (chunk start state)
    acc += tmp;
  }
  Pfinal[idx] = acc;
}

// ---------------------------------------------------------------------------
// Kernel 3: per-(batch,chunk) outputs.
//   Y = Q.P_start + trilInc(Q K^T) K ;  fro2 scan ;  out = tanh(gain*Y/(|P|+eps))*scale
// grid = B*NC workgroups, 256 threads (8 waves).
// ---------------------------------------------------------------------------
__global__ __launch_bounds__(256) void out_kernel(const float* __restrict__ qin,
                                                  const float* __restrict__ kin,
                                                  const float* __restrict__ Pst_all,
                                                  const float* __restrict__ gain,
                                                  const float* __restrict__ scale,
                                                  float* __restrict__ out) {
  const int bc = blockIdx.x;
  const float* Qc = qin + (size_t)bc * C * D;
  const float* Kc = kin + (size_t)bc * C * D;
  const float* P  = Pst_all + (size_t)bc * D * D;  // chunk-start state (incl. P_prev)
  float* Oc       = out + (size_t)bc * C * D;

  __shared__ float sK[C][D];       // 64 KB
  __shared__ float sA[C][C];       // 16 KB : trilInc(Q K^T)
  __shared__ float g1[C];          // k_t^T P k_t
  __shared__ float g2s[C];         // sum_{s<t} G[t,s]^2
  __shared__ float gdiag[C];       // G[t,t]^2
  __shared__ float dinv[C];        // 1/(sqrt(fro2_t)+eps)
  __shared__ float red[256];       // fro2_start reduction

  const int tid = threadIdx.x;
  for (int i = tid; i < C * D; i += 256) sK[i >> 8][i & 255] = Kc[i];
  if (tid < C) { g1[tid] = 0.f; g2s[tid] = 0.f; gdiag[tid] = 0.f; }

  // ||P_start||_F^2 partial sums
  float loc = 0.f;
  for (int i = tid; i < D * D; i += 256) { const float v = P[i]; loc = fmaf(v, v, loc); }
  red[tid] = loc;
  __syncthreads();
  for (int s = 128; s > 0; s >>= 1) {
    if (tid < s) red[tid] += red[tid + s];
    __syncthreads();
  }  // red[0] = fro2_start (kept live)

  const int wave = tid >> 5, lane = tid & 31;
  const int lo = lane & 15, hi = lane >> 4;

  // ---- A = Q K^T (masked, stored) and G = K K^T (folded into g2s/gdiag) ----
  // 16 A-tiles + 16 G-tiles (4x4 grid each), 4 tiles per wave.
  for (int t = 0; t < 4; ++t) {
    const int tau = wave * 4 + t;                  // wave-uniform
    const bool isA = (tau < 16);
    const int l16 = isA ? tau : tau - 16;
    const int mi = l16 & 3, ni = l16 >> 2;
    v8f acc = {};
    for (int k0 = 0; k0 < D; k0 += 4) {
      v2f a, b;
      if (isA) {
        const float* qp = Qc + (size_t)(mi * 16 + lo) * D + k0 + 2 * hi;
        a.x = qp[0]; a.y = qp[1];
      } else {
        a.x = sK[mi * 16 + lo][k0 + 2 * hi];
        a.y = sK[mi * 16 + lo][k0 + 2 * hi + 1];
      }
      b.x = sK[ni * 16 + lo][k0 + 2 * hi];         // B[k=d, n=s] = K[s, d]
      b.y = sK[ni * 16 + lo][k0 + 2 * hi + 1];
      acc = wmma_f32(a, b, acc);
    }
    for (int r = 0; r < 8; ++r) {
      const int trow = mi * 16 + r + 8 * hi, scol = ni * 16 + lo;
      const float v = acc[r];
      if (isA) {
        sA[trow][scol] = (scol <= trow) ? v : 0.f;  // inclusive causal mask
      } else {
        if (scol < trow)       atomicAdd(&g2s[trow], v * v);
        else if (scol == trow) gdiag[trow] = v * v;
      }
    }
  }
  __syncthreads();

  // ---- Y = Q.P + trilInc(A).K ; KP = K.P -> g1 ----
  // wave owns n-panels {2w, 2w+1}; all 4 m-blocks share each P B-fragment.
  v8f accY[2][4] = {};
  for (int np = 0; np < 2; ++np) {
    const int n = wave * 2 + np;
    v8f accKP[4] = {};
    for (int k0 = 0; k0 < D; k0 += 4) {
      v2f bp;
      bp.x = P[(size_t)(k0 + 2 * hi) * D + n * 16 + lo];
      bp.y = P[(size_t)(k0 + 2 * hi + 1) * D + n * 16 + lo];
      for (int m = 0; m < 4; ++m) {
        v2f aq, ak;
        const float* qp = Qc + (size_t)(m * 16 + lo) * D + k0 + 2 * hi;
        aq.x = qp[0]; aq.y = qp[1];
        ak.x = sK[m * 16 + lo][k0 + 2 * hi];
        ak.y = sK[m * 16 + lo][k0 + 2 * hi + 1];
        accY[np][m] = wmma_f32(aq, bp, accY[np][m]);
        accKP[m]    = wmma_f32(ak, bp, accKP[m]);
      }
    }
    // g1[t] += (K.P)[t,d] * K[t,d]  over this n-panel
    for (int m = 0; m < 4; ++m)
      for (int r = 0; r < 8; ++r) {
        const int trow = m * 16 + r + 8 * hi;
        atomicAdd(&g1[trow], accKP[m][r] * sK[trow][n * 16 + lo]);
      }
    // intra-chunk: Y += trilInc(A) . K
    for (int s0 = 0; s0 < C; s0 += 4) {
      v2f bk;
      bk.x = sK[s0 + 2 * hi][n * 16 + lo];         // B[k=s, n=d] = K[s, d]
      bk.y = sK[s0 + 2 * hi + 1][n * 16 + lo];
      for (int m = 0; m < 4; ++m) {
        v2f at;
        at.x = sA[m * 16 + lo][s0 + 2 * hi];
        at.y = sA[m * 16 + lo][s0 + 2 * hi + 1];
        accY[np][m] = wmma_f32(at, bk, accY[np][m]);
      }
    }
  }
  __syncthreads();  // all g1/g2s/gdiag complete

  // ---- sequential fro^2 scan over the chunk (C=64 tiny steps) ----
  if (tid == 0) {
    float run = red[0];
    for (int t = 0; t < C; ++t) {
      run += 2.f * (g1[t] + g2s[t]) + gdiag[t];
      dinv[t] = 1.f / (sqrtf(run) + EPS);
    }
  }
  __syncthreads();

  // ---- outputs: tanh(gain * Y * dinv) * scale ----
  for (int np = 0; np < 2; ++np) {
    const int n = wave * 2 + np;
    const int d = n * 16 + lo;
    const float gd = gain[d], sd = scale[d];
    for (int m = 0; m < 4; ++m)
      for (int r = 0; r < 8; ++r) {
        const int trow = m * 16 + r + 8 * hi;
        const float val = accY[np][m][r] * dinv[trow];
        Oc[(size_t)trow * D + d] = tanhf(gd * val) * sd;
      }
  }
}

// ---------------------------------------------------------------------------
extern "C" void kernel_launch(void* const* d_in, const int* in_sizes, int n_in,
                              void* d_out, int out_size, void* d_ws, size_t ws_size,
                              hipStream_t stream) {
  (void)in_sizes; (void)n_in; (void)out_size; (void)ws_size;
  const float* q     = (const float*)d_in[0];   // [B,L,D]
  const float* k     = (const float*)d_in[1];   // [B,L,D]
  const float* Pprev = (const float*)d_in[2];   // [B,D,D]
  const float* gain  = (const float*)d_in[3];   // [D]
  const float* scale = (const float*)d_in[4];   // [D]
  float* out = (float*)d_out;                   // [B,L,D] then [B,D,D]
  float* M   = (float*)d_ws;                    // B*NC*D*D floats = 32 MB scratch

  gram_kernel<<<B * NC, 256, 0, stream>>>(k, M);
  scan_kernel<<<(B * D * D) / 256, 256, 0, stream>>>(Pprev, M,
                                                     out + (size_t)B * L * D);
  out_kernel<<<B * NC, 256, 0, stream>>>(q, k, M, gain, scale, out);
}